// GraphAttentionLayer_27754078667330
// MI455X (gfx1250) — compile-verified
//
#include <hip/hip_runtime.h>
#include <hip/hip_bf16.h>

// Problem dims (fixed by setup_inputs)
// bs=16 nv=8 mf=10 mp=4 K=3 nr=12 hd=64 no=64 nt=8
// Out layout (flat f32): g_i[1024] | g_c[1024] | g_t[1024] | path_attn[5120] | attn_causal[10240]

typedef float v2f __attribute__((ext_vector_type(2)));
typedef float v8f __attribute__((ext_vector_type(8)));

// ---------------- JAX threefry2x32 (20 rounds, exact schedule) ----------------
__device__ __forceinline__ void threefry2x32(unsigned k0, unsigned k1,
                                             unsigned& x0, unsigned& x1) {
  const unsigned ks0 = k0, ks1 = k1, ks2 = k0 ^ k1 ^ 0x1BD11BDAu;
  x0 += ks0; x1 += ks1;
#define TF_R(r) { x0 += x1; x1 = (x1 << (r)) | (x1 >> (32 - (r))); x1 ^= x0; }
  TF_R(13) TF_R(15) TF_R(26) TF_R(6)   x0 += ks1; x1 += ks2 + 1u;
  TF_R(17) TF_R(29) TF_R(16) TF_R(24)  x0 += ks2; x1 += ks0 + 2u;
  TF_R(13) TF_R(15) TF_R(26) TF_R(6)   x0 += ks0; x1 += ks1 + 3u;
  TF_R(17) TF_R(29) TF_R(16) TF_R(24)  x0 += ks1; x1 += ks2 + 4u;
  TF_R(13) TF_R(15) TF_R(26) TF_R(6)   x0 += ks2; x1 += ks0 + 5u;
#undef TF_R
}

// jax.random.uniform(key(42), (5120,), minval=1e-10, maxval=1.0) element i
__device__ __forceinline__ float jax_uniform_5120(int i) {
  unsigned x0, x1; const bool first = i < 2560;
  if (first) { x0 = (unsigned)i; x1 = (unsigned)(i + 2560); }
  else       { x0 = (unsigned)(i - 2560); x1 = (unsigned)i; }
  threefry2x32(0u, 42u, x0, x1);
  const unsigned bits = first ? x0 : x1;
  float f = __uint_as_float((bits >> 9) | 0x3f800000u) - 1.0f;
  f = f * (1.0f - 1e-10f) + 1e-10f;
  return fmaxf(f, 1e-10f);
}

// ---------------- Kernel 1: Sdiag = PS(15360x12) @ Wdiag(12x64) via WMMA -----
// One wave per 16x16 output tile; K=12 covered by 3 chained V_WMMA_F32_16X16X4_F32.
// A layout (16x4 f32): lanes 0-15 -> row=lane, K=(0,1); lanes 16-31 -> row=lane-16, K=(2,3)
// B layout (4x16 f32): lanes 0-15 -> col=lane, K=(0,1); lanes 16-31 -> col=lane-16, K=(2,3)
// C/D layout: VGPR j -> M=j (lanes 0-15) / M=j+8 (lanes 16-31), N=lane&15
__global__ __launch_bounds__(256) void k_sdiag_wmma(
    const float* __restrict__ ps,      // (15360,12)
    const float* __restrict__ W,       // (12,64,64)
    float* __restrict__ sdiag)         // (15360,64)
{
  const int wave  = (blockIdx.x * 256 + (int)threadIdx.x) >> 5;  // 0..3839
  const int lane  = threadIdx.x & 31;
  const int mtile = wave >> 2;                                   // 0..959
  const int ntile = wave & 3;                                    // 0..3
  const int m0    = mtile << 4;
  const int lr    = lane & 15;
  const int half  = (lane < 16) ? 0 : 2;
  const int col   = (ntile << 4) + lr;

  v8f c = {};
#pragma unroll
  for (int kt = 0; kt < 3; ++kt) {
    const int ka = kt * 4 + half;
    v2f a, b;
    a.x = ps[(m0 + lr) * 12 + ka];
    a.y = ps[(m0 + lr) * 12 + ka + 1];
    b.x = W[(size_t)ka       * 4096 + col * 65];   // W[r,d,d]
    b.y = W[(size_t)(ka + 1) * 4096 + col * 65];
    c = __builtin_amdgcn_wmma_f32_16x16x4_f32(false, a, false, b,
                                              (short)0, c, false, false);
  }
#pragma unroll
  for (int j = 0; j < 8; ++j) {
    const int row = m0 + ((lane < 16) ? j : j + 8);
    sdiag[(size_t)row * 64 + col] = c[j];
  }
}

// ---------------- Kernel 2: per-(b,v,f) path stage ---------------------------
// mj[p,d] = (prod_k Sdiag) * feat ; gumbel-hard mask over p ; path_attn ;
// optional filtering ; contract with path_target -> mj2[b,v,f,t,d]
__global__ __launch_bounds__(256) void k_path(
    const float* __restrict__ sdiag, const float* __restrict__ feat,
    const float* __restrict__ h_time, const float* __restrict__ W_p,
    const float* __restrict__ pmlp_w, const float* __restrict__ pmlp_b,
    const float* __restrict__ pt, const int* __restrict__ path_filtering,
    float* __restrict__ mj2, float* __restrict__ out_path_attn)
{
  __shared__ float red[256];
  __shared__ float mjbuf[256];
  __shared__ float msgs[4];
  __shared__ float attnp[4];
  __shared__ float hsum_s;
  __shared__ int   sel_s;

  const int g = blockIdx.x;            // (b*8+v)*10+f, 0..1279
  const int b = g / 80;
  const int t = threadIdx.x;
  const int p = t >> 6, d = t & 63;

  const int rb = (g * 4 + p) * 3;      // flat row base into Sdiag
  float mj = sdiag[(size_t)(rb + 0) * 64 + d]
           * sdiag[(size_t)(rb + 1) * 64 + d]
           * sdiag[(size_t)(rb + 2) * 64 + d];
  mj *= feat[(size_t)g * 64 + d];

  // hsum = h_time[b,:] . W_p[0:64]
  red[t] = (t < 64) ? h_time[b * 64 + t] * W_p[t] : 0.0f;
  __syncthreads();
  for (int s = 32; s > 0; s >>= 1) { if (t < s) red[t] += red[t + s]; __syncthreads(); }
  if (t == 0) hsum_s = red[0];
  __syncthreads();

  // msg[p] = hsum + mj[p,:] . W_p[64:128]
  red[t] = mj * W_p[64 + d];
  __syncthreads();
  for (int s = 32; s > 0; s >>= 1) { if (d < s) red[t] += red[t + s]; __syncthreads(); }
  if (d == 0) msgs[p] = hsum_s + red[t];
  __syncthreads();

  // gumbel-hard: argmax_p (msg + g), first max wins (jnp.argmax tie rule)
  if (t == 0) {
    float best = -3.4e38f; int bi = 0;
    for (int pp = 0; pp < 4; ++pp) {
      const float u   = jax_uniform_5120(g * 4 + pp);
      const float gum = -logf(-logf(u));
      const float v   = msgs[pp] + gum;
      if (v > best) { best = v; bi = pp; }
    }
    sel_s = bi;
  }
  __syncthreads();

  // path_attn[p] = sigmoid(mask[p] * (mj . pmlp_w) + pmlp_b)
  red[t] = mj * pmlp_w[d];
  __syncthreads();
  for (int s = 32; s > 0; s >>= 1) { if (d < s) red[t] += red[t + s]; __syncthreads(); }
  if (d == 0) {
    const float dv = (p == sel_s) ? red[t] : 0.0f;
    const float a  = 1.0f / (1.0f + expf(-(dv + pmlp_b[0])));
    attnp[p] = a;
    out_path_attn[g * 4 + p] = a;
  }
  __syncthreads();

  if (*path_filtering) mj *= attnp[p];
  mjbuf[t] = mj;
  __syncthreads();

  // mj2[g,tt,dd] = sum_p pt[g,p,tt] * mj[p,dd]
  const float* ptg = pt + (size_t)g * 32;     // (p,tt) = 4x8
  for (int o = t; o < 512; o += 256) {
    const int tt = o >> 6, dd = o & 63;
    float acc = 0.0f;
#pragma unroll
    for (int pp = 0; pp < 4; ++pp)
      acc += ptg[pp * 8 + tt] * mjbuf[pp * 64 + dd];
    mj2[(size_t)g * 512 + (size_t)tt * 64 + dd] = acc;
  }
}

// ---------------- Kernel 3: per-(b,v,t) joint-impact + causal ----------------
__global__ __launch_bounds__(640) void k_joint_causal(
    const float* __restrict__ mj2, const float* __restrict__ h_time,
    const float* __restrict__ cmlp_w, const float* __restrict__ cmlp_b,
    const float* __restrict__ mlp_w, const float* __restrict__ mlp_b,
    const float* __restrict__ cthr, const int* __restrict__ joint_impact,
    float* __restrict__ out_gc, float* __restrict__ out_gt,
    float* __restrict__ out_ac)
{
  __shared__ float red[640];
  __shared__ float nbuf[640];
  __shared__ float mobuf[640];
  __shared__ float normf[10];
  __shared__ float dcs[10];
  __shared__ float ac[10];
  __shared__ float hc_s;

  const int q  = blockIdx.x;       // (b*8+v)*8+tt, 0..1023
  const int b  = q >> 6;
  const int bv = q >> 3;
  const int tt = q & 7;
  const int t  = threadIdx.x;      // 0..639
  const int f  = t / 64, d = t & 63;

  // Mp[f,d] = mj2[b,v,f,tt,d]
  const float mpv = mj2[((size_t)(bv * 10 + f) * 8 + tt) * 64 + d];
  float mo = mpv;

  if (*joint_impact) {
    // norm over d, clamp at 1e-12
    red[t] = mpv * mpv; __syncthreads();
    for (int s = 32; s > 0; s >>= 1) { if (d < s) red[t] += red[t + s]; __syncthreads(); }
    if (d == 0) normf[f] = fmaxf(sqrtf(red[t]), 1e-12f);
    __syncthreads();
    nbuf[t] = mpv / normf[f];     // flat (f*64+d) block, exactly the reshape view
    __syncthreads();
    // diag of reshape-cos: sum_d flat[f*64+d] * flat[d*10+f]
    red[t] = nbuf[t] * nbuf[d * 10 + f];
    __syncthreads();
    for (int s = 32; s > 0; s >>= 1) { if (d < s) red[t] += red[t + s]; __syncthreads(); }
    if (d == 0) dcs[f] = red[t];
    __syncthreads();
    mo = dcs[f] * mpv;            // scales Mp (not the normalized n)
  }

  // hdot = h_time[b,:] . cmlp_w[0:64]
  red[t] = (t < 64) ? h_time[b * 64 + t] * cmlp_w[t] : 0.0f;
  __syncthreads();
  for (int s = 32; s > 0; s >>= 1) { if (t < s) red[t] += red[t + s]; __syncthreads(); }
  if (t == 0) hc_s = red[0];
  __syncthreads();

  // score[f] = hdot + mo[f,:] . cmlp_w[64:128] + cb ; attn = sigmoid(score - thr)
  red[t] = mo * cmlp_w[64 + d];
  __syncthreads();
  for (int s = 32; s > 0; s >>= 1) { if (d < s) red[t] += red[t + s]; __syncthreads(); }
  if (d == 0) {
    const float sc = hc_s + red[t] + cmlp_b[0];
    const float a  = 1.0f / (1.0f + expf(-(sc - cthr[0])));
    ac[f] = a;
    out_ac[q * 10 + f] = a;
  }
  mobuf[t] = mo;
  __syncthreads();

  // g_c = (sum_f ac[f]*mo[f,:]) . mlp_w + mb ; g_t with (1-ac)
  if (t < 64) {
    float gcv = 0.0f, gtv = 0.0f;
#pragma unroll
    for (int ff = 0; ff < 10; ++ff) {
      const float m = mobuf[ff * 64 + t];
      const float a = ac[ff];
      gcv += a * m; gtv += (1.0f - a) * m;
    }
    red[t]  = gcv * mlp_w[t];
    nbuf[t] = gtv * mlp_w[t];
  }
  __syncthreads();
  for (int s = 32; s > 0; s >>= 1) {
    if (t < s) { red[t] += red[t + s]; nbuf[t] += nbuf[t + s]; }
    __syncthreads();
  }
  if (t == 0) {
    out_gc[q] = red[0]  + mlp_b[0];
    out_gt[q] = nbuf[0] + mlp_b[0];
  }
}

// ---------------- Kernel 4: intervention g_i = g_c + g_t[II[b]] --------------
__global__ __launch_bounds__(256) void k_intervene(
    const float* __restrict__ gc, const float* __restrict__ gt,
    float* __restrict__ gi)
{
  const int q = blockIdx.x * 256 + (int)threadIdx.x;
  if (q >= 1024) return;
  const int b = q >> 6;
  // jax.random.randint(key(7), (16,), 0, 16): span=16 (pow2) -> bits & 15
  unsigned x0, x1; const bool first = b < 8;
  if (first) { x0 = (unsigned)b; x1 = (unsigned)(b + 8); }
  else       { x0 = (unsigned)(b - 8); x1 = (unsigned)b; }
  threefry2x32(0u, 7u, x0, x1);
  const unsigned bits = first ? x0 : x1;
  int ii = (int)(bits & 15u);
  ii = (ii + (ii == b ? 1 : 0)) & 15;          // (II + (II==arange)) % 16
  gi[q] = gc[q] + gt[ii * 64 + (q & 63)];
}

// ---------------- Launch -----------------------------------------------------
extern "C" void kernel_launch(void* const* d_in, const int* in_sizes, int n_in,
                              void* d_out, int out_size, void* d_ws, size_t ws_size,
                              hipStream_t stream) {
  (void)in_sizes; (void)n_in; (void)out_size; (void)ws_size;
  const float* feat   = (const float*)d_in[0];   // (16,8,10,64)
  const float* h_time = (const float*)d_in[1];   // (16,64)
  /* d_in[2] path_index: shape-only in reference */
  const float* ps     = (const float*)d_in[3];   // (16,320,3,12) == (15360,12)
  const float* pt     = (const float*)d_in[4];   // (16,8,10,4,8)
  const float* W      = (const float*)d_in[5];   // (12,64,64)
  const float* W_p    = (const float*)d_in[6];   // (128,)
  const float* pmlp_w = (const float*)d_in[7];
  const float* pmlp_b = (const float*)d_in[8];
  const float* cmlp_w = (const float*)d_in[9];
  const float* cmlp_b = (const float*)d_in[10];
  const float* mlp_w  = (const float*)d_in[11];
  const float* mlp_b  = (const float*)d_in[12];
  const float* cthr   = (const float*)d_in[13];
  const int*   pf     = (const int*)d_in[14];
  const int*   ji     = (const int*)d_in[15];

  float* out   = (float*)d_out;
  float* sdiag = (float*)d_ws;              // 15360*64 f32 = 3.93 MB
  float* mj2   = sdiag + 15360 * 64;        // 1280*512 f32 = 2.62 MB

  k_sdiag_wmma <<<480,  256, 0, stream>>>(ps, W, sdiag);
  k_path       <<<1280, 256, 0, stream>>>(sdiag, feat, h_time, W_p, pmlp_w, pmlp_b,
                                          pt, pf, mj2, out + 3072);
  k_joint_causal<<<1024, 640, 0, stream>>>(mj2, h_time, cmlp_w, cmlp_b, mlp_w, mlp_b,
                                           cthr, ji, out + 1024, out + 2048, out + 8192);
  k_intervene  <<<4,    256, 0, stream>>>(out + 1024, out + 2048, out);
}